// PointConv_9783935500533
// MI455X (gfx1250) — compile-verified
//
#include <hip/hip_runtime.h>

// PointConv fused pipeline for gfx1250 (MI455X), f32 end-to-end using
// V_WMMA_F32_16X16X4_F32 for every GEMM stage (matches reference f32 numerics).

typedef float v2f __attribute__((ext_vector_type(2)));
typedef float v8f __attribute__((ext_vector_type(8)));

#define NPTS 4096
#define BS   8
#define KNN  32
#define CI   64
#define CO   64
#define HID  32
#define WN   16      // CMCO_CI
#define QT   16      // queries per workgroup tile (kernel 2)
#define CHROWS 128   // (query,k) rows per MLP chunk

static __device__ __forceinline__ float swishf(float x) {
  // x * sigmoid(x); single v_rcp_f32 instead of the IEEE div expansion
  // (v_div_scale + Newton chain) the compiler emits for '/'.
  return x * __builtin_amdgcn_rcpf(1.0f + __expf(-x));
}

static __device__ __forceinline__ v8f wmma4(v2f a, v2f b, v8f c) {
  // D(16x16,f32) = A(16x4,f32) * B(4x16,f32) + C
  return __builtin_amdgcn_wmma_f32_16x16x4_f32(false, a, false, b, (short)0, c,
                                               false, false);
}

// ---------------------------------------------------------------------------
// Kernel 1: brute-force KNN (top-32 by squared distance).
// One thread per query; candidates staged through LDS in 256-point chunks.
// Top-32 kept unsorted in registers (replace current max, rescan for new max);
// the downstream reduction sums over K, so set membership is all that matters.
// ---------------------------------------------------------------------------
__global__ __launch_bounds__(256) void knn_kernel(
    const float* __restrict__ xyz, const unsigned char* __restrict__ mask,
    int* __restrict__ idx_out) {
  __shared__ float sCx[256], sCy[256], sCz[256];
  __shared__ unsigned char sM[256];

  const int t = threadIdx.x;
  const int b = blockIdx.x >> 4;                 // 16 blocks per batch
  const int m = ((blockIdx.x & 15) << 8) + t;    // query index
  const float* xb = xyz + (size_t)b * NPTS * 3;

  const float qx = xb[m * 3 + 0];
  const float qy = xb[m * 3 + 1];
  const float qz = xb[m * 3 + 2];

  float best[KNN];
  int bidx[KNN];
#pragma unroll
  for (int s = 0; s < KNN; ++s) { best[s] = 3.0e38f; bidx[s] = 0; }
  float maxv = 3.0e38f;
  int maxslot = 0;

  for (int ch = 0; ch < NPTS / 256; ++ch) {
    const int j = ch * 256 + t;
    sCx[t] = xb[j * 3 + 0];
    sCy[t] = xb[j * 3 + 1];
    sCz[t] = xb[j * 3 + 2];
    sM[t] = mask[(size_t)b * NPTS + j];
    __syncthreads();
    for (int jj = 0; jj < 256; ++jj) {
      const float dx = qx - sCx[jj];
      const float dy = qy - sCy[jj];
      const float dz = qz - sCz[jj];
      float d2 = dx * dx + dy * dy + dz * dz;
      if (!sM[jj]) d2 = 1.0e10f;   // reference masks invalid points to 1e10
      if (d2 < maxv) {
        const int cand = ch * 256 + jj;
#pragma unroll
        for (int s = 0; s < KNN; ++s)
          if (s == maxslot) { best[s] = d2; bidx[s] = cand; }
        maxv = -1.0f;
#pragma unroll
        for (int s = 0; s < KNN; ++s)
          if (best[s] > maxv) { maxv = best[s]; maxslot = s; }
      }
    }
    __syncthreads();
  }

  int* op = idx_out + ((size_t)b * NPTS + m) * KNN;
#pragma unroll
  for (int s = 0; s < KNN; ++s) op[s] = bidx[s];
}

// ---------------------------------------------------------------------------
// Kernel 2: fused gather + WeightNet MLP + neighbor einsum + final linear.
// One 256-thread workgroup (8 wave32) per 16-query tile.
// ---------------------------------------------------------------------------
struct TileLDS {
  float dl[CHROWS][4];        // deltas chunk, K padded 3->4
  float h1[CHROWS][HID];      // layer-1 activations
  float h2[CHROWS][HID];      // layer-2 activations
  float wgt[QT * KNN][WN];    // weightnet outputs (all 512 rows)
  float p[QT][CI * WN];       // partial (bmcw) flattened c*16+w
  float o2[2][QT][CO];        // final GEMM K-split partials
  int nidx[QT][KNN];          // neighbor indices
  float w1[4][HID];           // W1, K row 3 zero-padded
  float w2[HID][HID];
  float w3[HID][WN];
  float bb1[HID], bb2[HID], bb3[WN], bbl[CO];
  float qx[QT][3];
  unsigned char qm[QT];
};

__global__ __launch_bounds__(256) void pointconv_fused_kernel(
    const float* __restrict__ xyz, const float* __restrict__ vals,
    const unsigned char* __restrict__ mask, const int* __restrict__ idxws,
    const float* __restrict__ W1, const float* __restrict__ b1,
    const float* __restrict__ W2, const float* __restrict__ b2,
    const float* __restrict__ W3, const float* __restrict__ b3,
    const float* __restrict__ Wl, const float* __restrict__ bl,
    float* __restrict__ outp) {
  __shared__ TileLDS L;

  const int t = threadIdx.x;
  const int wv = t >> 5;        // wave id 0..7
  const int l = t & 31;         // lane
  const int mrow = l & 15;      // M (or N) index within fragment
  const int half = l >> 4;      // selects K pair in A/B fragments

  const int b = blockIdx.x >> 8;             // 256 tiles per batch
  const int m0 = (blockIdx.x & 255) * QT;

  const float* xb = xyz + (size_t)b * NPTS * 3;
  const float* vb = vals + (size_t)b * NPTS * CI;

  // ---- Phase 0: stage indices, query xyz/mask, and all weightnet weights ----
  {
    const int* gidx = idxws + ((size_t)b * NPTS + m0) * KNN;
    int* ni = &L.nidx[0][0];
    for (int f = t; f < QT * KNN; f += 256) ni[f] = gidx[f];
    const float* qxyz = xyz + ((size_t)b * NPTS + m0) * 3;
    float* qf = &L.qx[0][0];
    for (int f = t; f < QT * 3; f += 256) qf[f] = qxyz[f];
    if (t < QT) L.qm[t] = mask[(size_t)b * NPTS + m0 + t];
    for (int f = t; f < 4 * HID; f += 256) {
      const int r = f / HID, c = f % HID;
      L.w1[r][c] = (r < 3) ? W1[r * HID + c] : 0.0f;
    }
    for (int f = t; f < HID * HID; f += 256) (&L.w2[0][0])[f] = W2[f];
    for (int f = t; f < HID * WN; f += 256) (&L.w3[0][0])[f] = W3[f];
    if (t < HID) { L.bb1[t] = b1[t]; L.bb2[t] = b2[t]; }
    if (t < WN) L.bb3[t] = b3[t];
    if (t < CO) L.bbl[t] = bl[t];
    // Warm Wl (256 KB, re-read by every workgroup) into the cache hierarchy
    // while the MLP phases run: one global_prefetch_b8 per 128B line.
#pragma unroll
    for (int pf = 0; pf < 8; ++pf) {
      __builtin_prefetch(
          (const char*)Wl + ((size_t)pf * 256 + t) * 128, 0, 3);
    }
  }
  __syncthreads();

  // ---- MLP over 512 (query,k) rows in 4 chunks of 128 rows ----
  for (int cc = 0; cc < (QT * KNN) / CHROWS; ++cc) {
    // deltas for this chunk
    if (t < CHROWS) {
      const int r = t;
      const int gq = (cc * CHROWS + r) >> 5;   // query in tile
      const int k = r & (KNN - 1);
      const int j = L.nidx[gq][k];
      L.dl[r][0] = L.qx[gq][0] - xb[j * 3 + 0];
      L.dl[r][1] = L.qx[gq][1] - xb[j * 3 + 1];
      L.dl[r][2] = L.qx[gq][2] - xb[j * 3 + 2];
      L.dl[r][3] = 0.0f;
    }
    __syncthreads();

    const int r0 = wv * 16;   // each wave owns one 16-row tile of the chunk

    // layer 1: (16x4) @ (4x32), single K step (row 3 zero)
    {
      v2f a = {L.dl[r0 + mrow][2 * half], L.dl[r0 + mrow][2 * half + 1]};
#pragma unroll
      for (int nt = 0; nt < 2; ++nt) {
        v2f bf = {L.w1[2 * half][nt * 16 + mrow],
                  L.w1[2 * half + 1][nt * 16 + mrow]};
        v8f acc = {};
        acc = wmma4(a, bf, acc);
#pragma unroll
        for (int i = 0; i < 8; ++i) {
          const int row = r0 + i + 8 * half;
          const int col = nt * 16 + mrow;
          L.h1[row][col] = swishf(acc[i] + L.bb1[col]);
        }
      }
    }
    __syncthreads();

    // layer 2: (16x32) @ (32x32), 8 K steps x 2 N tiles
    {
#pragma unroll
      for (int nt = 0; nt < 2; ++nt) {
        v8f acc = {};
#pragma unroll
        for (int kk = 0; kk < 8; ++kk) {
          const int k0 = kk * 4 + 2 * half;
          v2f a = {L.h1[r0 + mrow][k0], L.h1[r0 + mrow][k0 + 1]};
          v2f bf = {L.w2[k0][nt * 16 + mrow], L.w2[k0 + 1][nt * 16 + mrow]};
          acc = wmma4(a, bf, acc);
        }
#pragma unroll
        for (int i = 0; i < 8; ++i) {
          const int row = r0 + i + 8 * half;
          const int col = nt * 16 + mrow;
          L.h2[row][col] = swishf(acc[i] + L.bb2[col]);
        }
      }
    }
    __syncthreads();

    // layer 3: (16x32) @ (32x16), 8 K steps
    {
      v8f acc = {};
#pragma unroll
      for (int kk = 0; kk < 8; ++kk) {
        const int k0 = kk * 4 + 2 * half;
        v2f a = {L.h2[r0 + mrow][k0], L.h2[r0 + mrow][k0 + 1]};
        v2f bf = {L.w3[k0][mrow], L.w3[k0 + 1][mrow]};
        acc = wmma4(a, bf, acc);
      }
#pragma unroll
      for (int i = 0; i < 8; ++i) {
        const int row = cc * CHROWS + r0 + i + 8 * half;
        L.wgt[row][mrow] = swishf(acc[i] + L.bb3[mrow]);
      }
    }
    __syncthreads();
  }

  // ---- einsum: per query, partial(64x16) = vals^T(64x32) @ wgt(32x16) ----
  // 64 tasks = 16 queries x 4 c-tiles; vals A-fragments gathered from global
  // (lanes 0..15 read consecutive c for one neighbor row -> coalesced, L2 hot).
  for (int task = wv; task < QT * 4; task += 8) {
    const int q = task >> 2;
    const int ct = task & 3;
    const int c0 = ct * 16 + mrow;
    v8f acc = {};
#pragma unroll
    for (int kk = 0; kk < 8; ++kk) {
      const int k0 = kk * 4 + 2 * half;
      const int j0 = L.nidx[q][k0];
      const int j1 = L.nidx[q][k0 + 1];
      v2f a = {vb[(size_t)j0 * CI + c0], vb[(size_t)j1 * CI + c0]};
      v2f bf = {L.wgt[q * KNN + k0][mrow], L.wgt[q * KNN + k0 + 1][mrow]};
      acc = wmma4(a, bf, acc);
    }
#pragma unroll
    for (int i = 0; i < 8; ++i) {
      const int crow = ct * 16 + i + 8 * half;
      L.p[q][crow * WN + mrow] = acc[i];     // flatten as c*16 + w (reference)
    }
  }
  __syncthreads();

  // ---- final linear: out(16x64) = P(16x1024) @ Wl(1024x64) ----
  // 8 waves = 4 N tiles x 2 K halves; K halves reduced through LDS.
  {
    const int nt = wv & 3;
    const int kh = wv >> 2;
    v8f acc = {};
#pragma unroll 4
    for (int kk = 0; kk < 128; ++kk) {
      const int k0 = kh * 512 + kk * 4 + 2 * half;
      v2f a = {L.p[mrow][k0], L.p[mrow][k0 + 1]};
      v2f bf = {Wl[(size_t)k0 * CO + nt * 16 + mrow],
                Wl[(size_t)(k0 + 1) * CO + nt * 16 + mrow]};
      acc = wmma4(a, bf, acc);
    }
#pragma unroll
    for (int i = 0; i < 8; ++i) {
      const int q = i + 8 * half;
      L.o2[kh][q][nt * 16 + mrow] = acc[i];
    }
  }
  __syncthreads();

  // ---- reduce K halves, add bias, apply output mask, store ----
  for (int e = t; e < QT * CO; e += 256) {
    const int q = e >> 6;
    const int c = e & 63;
    float v = L.o2[0][q][c] + L.o2[1][q][c] + L.bbl[c];
    if (!L.qm[q]) v = 0.0f;
    outp[((size_t)b * NPTS + m0 + q) * CO + c] = v;
  }
}

// ---------------------------------------------------------------------------
extern "C" void kernel_launch(void* const* d_in, const int* in_sizes, int n_in,
                              void* d_out, int out_size, void* d_ws,
                              size_t ws_size, hipStream_t stream) {
  const float* xyz = (const float*)d_in[0];
  const float* vals = (const float*)d_in[1];
  const unsigned char* mask = (const unsigned char*)d_in[2];  // jnp bool = 1B
  const float* W1 = (const float*)d_in[3];
  const float* b1 = (const float*)d_in[4];
  const float* W2 = (const float*)d_in[5];
  const float* b2 = (const float*)d_in[6];
  const float* W3 = (const float*)d_in[7];
  const float* b3 = (const float*)d_in[8];
  const float* Wl = (const float*)d_in[9];
  const float* bl = (const float*)d_in[10];
  float* out = (float*)d_out;
  int* idxws = (int*)d_ws;   // bs*n*K ints = 4 MB scratch

  knn_kernel<<<BS * (NPTS / 256), 256, 0, stream>>>(xyz, mask, idxws);
  pointconv_fused_kernel<<<BS * (NPTS / QT), 256, 0, stream>>>(
      xyz, vals, mask, idxws, W1, b1, W2, b2, W3, b3, Wl, bl, out);
}